// TreeMessagePasser_69750268887206
// MI455X (gfx1250) — compile-verified
//
#include <hip/hip_runtime.h>
#include <hip/hip_bf16.h>
#include <stdint.h>

// Problem constants (from the reference file).
#define N_NODES 2048
#define RDIM    32

typedef __attribute__((ext_vector_type(2))) float v2f;
typedef __attribute__((ext_vector_type(4))) float f4;
typedef __attribute__((ext_vector_type(8))) float v8f;

// ---------------------------------------------------------------------------
// Kernel 1: zero the ancestor matrix A (N x N f32) and extract parent[] from
// the dense bool children matrix (children[p*N+c] != 0  =>  parent[c] = p).
// Each child has exactly one parent, so the parent[] writes are race-free.
// ---------------------------------------------------------------------------
__global__ void k_prep(const uint8_t* __restrict__ ch,
                       float* __restrict__ A,
                       int* __restrict__ parent) {
  size_t idx    = (size_t)blockIdx.x * blockDim.x + threadIdx.x;
  size_t stride = (size_t)gridDim.x * blockDim.x;
  const size_t total = (size_t)N_NODES * N_NODES;
  for (; idx < total; idx += stride) {
    A[idx] = 0.0f;
    if (ch[idx]) {
      int p = (int)(idx / N_NODES);
      int c = (int)(idx % N_NODES);
      parent[c] = p;
    }
  }
}

// ---------------------------------------------------------------------------
// Kernel 2: build A. Column d gets 1.0 at every row on d's ancestor chain
// (including d itself). Then A[i][d] = 1 iff d is in subtree(i), and
// final = A * features (the upward additive pass telescopes to subtree sums).
// ---------------------------------------------------------------------------
__global__ void k_build_A(const int* __restrict__ parent,
                          float* __restrict__ A) {
  int d = blockIdx.x * blockDim.x + threadIdx.x;
  if (d >= N_NODES) return;
  int i = d;
  while (true) {
    A[(size_t)i * N_NODES + d] = 1.0f;
    if (i == N_NODES - 1) break;   // root reached
    i = parent[i];
  }
}

// ---------------------------------------------------------------------------
// Kernel 3: final = A (2048x2048) * features (2048x32) in f32 via
// V_WMMA_F32_16X16X4_F32. One wave owns a 16(M) x 32(N) output tile:
// two 16x16 f32 accumulators, K swept in steps of 4.
//
// Operand layouts (ISA 7.12.2, wave32):
//   A 16x4 : lane m=lane%16 is row M; half h=lane/16; VGPR v holds K=2h+v.
//   B 4x16 : lane m is column N; VGPR v holds K=2h+v (rows striped by lane).
//   C/D    : VGPR v, half h -> M = v + 8h, N = lane%16.
// Grid 16 blocks x 8 waves = 128 M-tiles = 2048 rows exactly; no divergence,
// EXEC is all-ones around every WMMA as required.
// ---------------------------------------------------------------------------
__global__ void k_final_gemm(const float* __restrict__ A,
                             const float* __restrict__ F,
                             float* __restrict__ out) {
  const int lane  = threadIdx.x & 31;
  const int wave  = threadIdx.x >> 5;
  const int m     = lane & 15;
  const int h     = lane >> 4;
  const int mtile = blockIdx.x * (blockDim.x >> 5) + wave;
  const int mbase = mtile * 16;

  v8f c0 = {};  // columns 0..15
  v8f c1 = {};  // columns 16..31

  const float* arow = A + (size_t)(mbase + m) * N_NODES;

  for (int kb = 0; kb < N_NODES; kb += 4) {
    const int kk = kb + 2 * h;
    v2f a, b0, b1;
    a.x  = arow[kk];
    a.y  = arow[kk + 1];
    b0.x = F[kk * RDIM + m];
    b0.y = F[(kk + 1) * RDIM + m];
    b1.x = F[kk * RDIM + m + 16];
    b1.y = F[(kk + 1) * RDIM + m + 16];
    // 8 args: (neg_a, A, neg_b, B, c_mod, C, reuse_a, reuse_b)
    c0 = __builtin_amdgcn_wmma_f32_16x16x4_f32(false, a, false, b0,
                                               (short)0, c0, false, false);
    c1 = __builtin_amdgcn_wmma_f32_16x16x4_f32(false, a, false, b1,
                                               (short)0, c1, false, false);
  }

#pragma unroll
  for (int v = 0; v < 8; ++v) {
    const int row = mbase + v + 8 * h;
    out[row * RDIM + m]      = c0[v];
    out[row * RDIM + m + 16] = c1[v];
  }
}

// ---------------------------------------------------------------------------
// Kernel 4: trajectory writer — the real bandwidth work (512 MB of stores,
// ~22 us floor at 23.3 TB/s). Row t is: final[j] for j<=t, init[j] for j>t.
// final/init are 256 KB each -> L2-resident across all 2048 blocks; the
// write-once trajectory goes out through non-temporal 128-bit stores so it
// does not pollute L2.
// ---------------------------------------------------------------------------
__global__ void k_traj(const float* __restrict__ fin,
                       const float* __restrict__ init,
                       float* __restrict__ traj) {
  const int t = blockIdx.x;
  const f4* __restrict__ f4fin  = (const f4*)fin;
  const f4* __restrict__ f4init = (const f4*)init;
  const int row_f4 = N_NODES * RDIM / 4;                 // 16384 float4 per row
  f4* __restrict__ dst = (f4*)traj + (size_t)t * row_f4;

  for (int k = threadIdx.x; k < row_f4; k += blockDim.x) {
    const int j = k >> 3;                                // 8 float4 per node (R=32)
    const f4* __restrict__ src = (j <= t) ? f4fin : f4init;
    __builtin_nontemporal_store(src[k], dst + k);
  }
}

// ---------------------------------------------------------------------------
// d_in[0] = representations (N*R f32), d_in[1] = features (N*D f32),
// d_in[2] = children (N*N bool, 1 byte/elem as stored by jnp).
// d_out   = [final (N*R f32)][trajectory (N*N*R f32)].
// d_ws    : parent[] at offset 0 (8 KB), A at offset 8 KB (16 MB).
// ---------------------------------------------------------------------------
extern "C" void kernel_launch(void* const* d_in, const int* in_sizes, int n_in,
                              void* d_out, int out_size, void* d_ws, size_t ws_size,
                              hipStream_t stream) {
  (void)in_sizes; (void)n_in; (void)out_size; (void)ws_size;

  const float*   reps     = (const float*)d_in[0];
  const float*   feats    = (const float*)d_in[1];
  const uint8_t* children = (const uint8_t*)d_in[2];

  float* out_final = (float*)d_out;
  float* out_traj  = out_final + N_NODES * RDIM;

  int*   parent = (int*)d_ws;
  float* A      = (float*)((char*)d_ws + 8192);

  k_prep<<<4096, 256, 0, stream>>>(children, A, parent);
  k_build_A<<<N_NODES / 256, 256, 0, stream>>>(parent, A);
  k_final_gemm<<<16, 256, 0, stream>>>(A, feats, out_final);
  k_traj<<<N_NODES, 256, 0, stream>>>(out_final, reps, out_traj);
}